// HANLayer_81853486728019
// MI455X (gfx1250) — compile-verified
//
#include <hip/hip_runtime.h>
#include <math.h>

// ---------------- problem constants (match reference) ----------------
#define NN    100000     // N_NODES
#define NE    500000     // N_EDGES
#define NP    3          // N_PATHS
#define DIM   128        // D == HID
#define KL    10         // K_LAYERS
#define ND    (NN * DIM)         // 12,800,000 floats per path-embedding
#define ND4   (ND / 4)           // float4 count
#define TILES (NN / 16)          // 6250 exact (no tail)

typedef __attribute__((ext_vector_type(2))) float v2f;
typedef __attribute__((ext_vector_type(8))) float v8f;

// Guaranteed hardware f32 atomic add: GLOBAL_ATOMIC_ADD_F32, no-return form
// (tracked by STOREcnt; kernel-end implicit wait covers completion). This
// avoids any chance of a CAS-loop expansion in the bandwidth-critical
// scatter phase (~25 GB of L2-resident RMW traffic).
__device__ __forceinline__ void atomAddF(float* p, float v) {
  asm volatile("global_atomic_add_f32 %0, %1, off"
               :
               : "v"(p), "v"(v)
               : "memory");
}

// ---------------- utility kernels ----------------
__global__ __launch_bounds__(256) void fill_kernel(float* __restrict__ p, int n, float v) {
  int i = blockIdx.x * 256 + threadIdx.x;
  if (i < n) p[i] = v;
}

__global__ __launch_bounds__(256) void degree_kernel(const int* __restrict__ src,
                                                     const int* __restrict__ dst,
                                                     float* __restrict__ outdeg,
                                                     float* __restrict__ indeg, int E) {
  int e = blockIdx.x * 256 + threadIdx.x;
  if (e < E) {
    atomAddF(&outdeg[src[e]], 1.0f);
    atomAddF(&indeg[dst[e]], 1.0f);
  }
}

__global__ __launch_bounds__(256) void norm_kernel(float* __restrict__ ns,
                                                   float* __restrict__ nd, int n) {
  int i = blockIdx.x * 256 + threadIdx.x;
  if (i < n) {
    ns[i] = rsqrtf(fmaxf(ns[i], 1.0f));   // clip(deg,1)^-0.5
    nd[i] = rsqrtf(fmaxf(nd[i], 1.0f));
  }
}

// z_p = h0 ; xs = h0 * norm_s   (xs is the pre-scaled gather source)
__global__ __launch_bounds__(256) void init_kernel(const float4* __restrict__ h4,
                                                   const float* __restrict__ ns,
                                                   float4* __restrict__ z4,
                                                   float4* __restrict__ xs4, int n4) {
  int i = blockIdx.x * 256 + threadIdx.x;
  if (i >= n4) return;
  float4 hv = h4[i];
  z4[i] = hv;
  float s = ns[i >> 5];                    // 32 float4 per 128-float row
  float4 o; o.x = hv.x * s; o.y = hv.y * s; o.z = hv.z * s; o.w = hv.w * s;
  xs4[i] = o;
}

// One wave per edge: lane l gathers float4 l of xs[src] and scatter-adds to agg[dst].
// Working set (xs + agg, ~100 MB) is L2-resident on the 192 MB L2; only the
// edge stream (4 MB/path per pass) comes from HBM.
__global__ __launch_bounds__(256) void scatter_kernel(const int* __restrict__ src,
                                                      const int* __restrict__ dst,
                                                      const float4* __restrict__ xs4,
                                                      float* __restrict__ agg, int E) {
  int w    = (blockIdx.x * 256 + threadIdx.x) >> 5;
  int lane = threadIdx.x & 31;
  if (w >= E) return;
  int s = src[w];
  int d = dst[w];
  float4 v = xs4[(size_t)s * 32 + lane];
  float* a = agg + (size_t)d * DIM + lane * 4;
  atomAddF(a + 0, v.x);
  atomAddF(a + 1, v.y);
  atomAddF(a + 2, v.z);
  atomAddF(a + 3, v.w);
}

// x = 0.9*agg*norm_d + 0.1*h0 ; write x into z_p and xs = x*norm_s for next
// layer. Also re-zeros agg in-place (cache-hot line) so the next iteration
// needs no separate fill pass.
__global__ __launch_bounds__(256) void combine_kernel(float4* __restrict__ agg4,
                                                      const float4* __restrict__ h4,
                                                      const float* __restrict__ ns,
                                                      const float* __restrict__ ndg,
                                                      float4* __restrict__ z4,
                                                      float4* __restrict__ xs4, int n4) {
  int i = blockIdx.x * 256 + threadIdx.x;
  if (i >= n4) return;
  int node = i >> 5;
  float dn = ndg[node], sn = ns[node];
  float4 a = agg4[i], hv = h4[i];
  float4 zero; zero.x = 0.f; zero.y = 0.f; zero.z = 0.f; zero.w = 0.f;
  agg4[i] = zero;                          // fused zeroing for next iteration
  float4 x;
  x.x = 0.9f * a.x * dn + 0.1f * hv.x;
  x.y = 0.9f * a.y * dn + 0.1f * hv.y;
  x.z = 0.9f * a.z * dn + 0.1f * hv.z;
  x.w = 0.9f * a.w * dn + 0.1f * hv.w;
  z4[i] = x;
  float4 o; o.x = x.x * sn; o.y = x.y * sn; o.z = x.z * sn; o.w = x.w * sn;
  xs4[i] = o;
}

// ---------------- attention head: WMMA f32 16x16x4 ----------------
// Per wave: one (path, 16-node tile). S = z_tile(16x128) @ W1(128x128) + b1,
// score contribution = sum over tile of tanh(S)*w2. Only the SUM over all
// nodes is needed (w.mean), so each lane folds its own C-fragment elements
// and the wave reduces with shfl_xor; lane 0 atomically adds to ssum[p].
__global__ __launch_bounds__(256) void attn_kernel(const float* __restrict__ z,
                                                   const float* __restrict__ W1,
                                                   const float* __restrict__ b1,
                                                   const float* __restrict__ w2,
                                                   float* __restrict__ ssum) {
  __shared__ float sW1[DIM * DIM];   // 64 KB of the 320 KB WGP LDS
  __shared__ float sB1[DIM];
  __shared__ float sW2[DIM];

  // cooperative stage of W1 / b1 / w2 into LDS
  const float4* W14 = (const float4*)W1;
  float4* sW14 = (float4*)sW1;
  for (int i = threadIdx.x; i < (DIM * DIM) / 4; i += 256) sW14[i] = W14[i];
  if (threadIdx.x < DIM) { sB1[threadIdx.x] = b1[threadIdx.x]; sW2[threadIdx.x] = w2[threadIdx.x]; }
  __syncthreads();

  int wave = threadIdx.x >> 5;
  int lane = threadIdx.x & 31;
  int t    = blockIdx.x * 8 + wave;          // global tile id (wave-uniform)
  if (t >= NP * TILES) return;               // whole-wave exit: EXEC stays all-1 for WMMA
  int p  = t / TILES;
  int r0 = (t % TILES) * 16;                 // NN % 16 == 0 -> no tail masking

  const float* zp = z + (size_t)p * ND;
  int m  = lane & 15;                        // N index of C fragment / M index of A fragment
  int hi = lane >> 4;                        // 0: K0/K1 half, 1: K2/K3 half
  // A fragment source: row (r0+m), consecutive K pair starting at even k0.
  const float2* arow = (const float2*)(zp + (size_t)(r0 + m) * DIM);

  float local = 0.0f;
  for (int nt = 0; nt < 8; ++nt) {
    int n0 = nt * 16;
    float bv = sB1[n0 + m];
    v8f c = {bv, bv, bv, bv, bv, bv, bv, bv};      // C[M][N] init = b1[N]
    for (int kb = 0; kb < 32; ++kb) {
      int k0 = kb * 4 + hi * 2;                    // even
      float2 ap = arow[kb * 2 + hi];               // A[m][k0], A[m][k0+1]
      v2f a; a.x = ap.x; a.y = ap.y;
      v2f b;                                       // B[k0][n0+m], B[k0+1][n0+m]
      b.x = sW1[(size_t)k0 * DIM + n0 + m];
      b.y = sW1[(size_t)(k0 + 1) * DIM + n0 + m];
      c = __builtin_amdgcn_wmma_f32_16x16x4_f32(false, a, false, b,
                                                (short)0, c, false, false);
    }
    float wv = sW2[n0 + m];
    #pragma unroll
    for (int v = 0; v < 8; ++v) local += tanhf(c[v]) * wv;
  }
  // wave32 reduction
  for (int off = 16; off > 0; off >>= 1) local += __shfl_xor(local, off, 32);
  if (lane == 0) atomAddF(&ssum[p], local);
}

// softmax over the 3 per-path means -> beta stored at ssum[4..6]
__global__ void beta_kernel(float* __restrict__ ssum) {
  float w0 = ssum[0] / (float)NN;
  float w1 = ssum[1] / (float)NN;
  float w2v = ssum[2] / (float)NN;
  float mx = fmaxf(w0, fmaxf(w1, w2v));
  float e0 = expf(w0 - mx), e1 = expf(w1 - mx), e2 = expf(w2v - mx);
  float inv = 1.0f / (e0 + e1 + e2);
  ssum[4] = e0 * inv; ssum[5] = e1 * inv; ssum[6] = e2 * inv;
}

__global__ __launch_bounds__(256) void final_kernel(const float4* __restrict__ z4,
                                                    const float* __restrict__ beta,
                                                    float4* __restrict__ out4, int n4) {
  int i = blockIdx.x * 256 + threadIdx.x;
  if (i >= n4) return;
  float b0 = beta[4], b1v = beta[5], b2 = beta[6];
  float4 a = z4[i], b = z4[i + (size_t)ND4], c = z4[i + (size_t)2 * ND4];
  float4 o;
  o.x = b0 * a.x + b1v * b.x + b2 * c.x;
  o.y = b0 * a.y + b1v * b.y + b2 * c.y;
  o.z = b0 * a.z + b1v * b.z + b2 * c.z;
  o.w = b0 * a.w + b1v * b.w + b2 * c.w;
  out4[i] = o;
}

// ---------------- host launcher ----------------
extern "C" void kernel_launch(void* const* d_in, const int* in_sizes, int n_in,
                              void* d_out, int out_size, void* d_ws, size_t ws_size,
                              hipStream_t stream) {
  (void)in_sizes; (void)n_in; (void)out_size; (void)ws_size;
  const float* h     = (const float*)d_in[0];
  const int*   edges = (const int*)d_in[1];   // (3,2,NE) int32
  const float* W1    = (const float*)d_in[2];
  const float* b1    = (const float*)d_in[3];
  const float* w2    = (const float*)d_in[4];
  float* out = (float*)d_out;
  float* ws  = (float*)d_ws;

  // workspace layout (floats): z[3*ND] | xs[ND] | norm_s[NN] | norm_d[NN] | ssum[8]
  float* z      = ws;
  float* xs     = z + (size_t)NP * ND;
  float* norm_s = xs + ND;
  float* norm_d = norm_s + NN;
  float* ssum   = norm_d + NN;
  float* agg    = out;                        // reuse d_out (exactly ND floats) as scatter accum

  const int B = 256;
  for (int p = 0; p < NP; ++p) {
    const int* src = edges + (size_t)p * 2 * NE;
    const int* dst = src + NE;
    float* zp = z + (size_t)p * ND;

    fill_kernel<<<(2 * NN + B - 1) / B, B, 0, stream>>>(norm_s, 2 * NN, 0.0f);
    degree_kernel<<<(NE + B - 1) / B, B, 0, stream>>>(src, dst, norm_s, norm_d, NE);
    norm_kernel<<<(NN + B - 1) / B, B, 0, stream>>>(norm_s, norm_d, NN);
    init_kernel<<<(ND4 + B - 1) / B, B, 0, stream>>>((const float4*)h, norm_s,
                                                     (float4*)zp, (float4*)xs, ND4);
    // agg (= d_out) holds stale data from the previous launch's output:
    // zero it once per path; combine_kernel re-zeros it every iteration.
    fill_kernel<<<(ND + B - 1) / B, B, 0, stream>>>(agg, ND, 0.0f);
    for (int k = 0; k < KL; ++k) {
      scatter_kernel<<<(NE * 32 + B - 1) / B, B, 0, stream>>>(src, dst, (const float4*)xs, agg, NE);
      combine_kernel<<<(ND4 + B - 1) / B, B, 0, stream>>>((float4*)agg, (const float4*)h,
                                                          norm_s, norm_d,
                                                          (float4*)zp, (float4*)xs, ND4);
    }
  }

  fill_kernel<<<1, 8, 0, stream>>>(ssum, 8, 0.0f);
  attn_kernel<<<(NP * TILES + 7) / 8, B, 0, stream>>>(z, W1, b1, w2, ssum);
  beta_kernel<<<1, 1, 0, stream>>>(ssum);
  final_kernel<<<(ND4 + B - 1) / B, B, 0, stream>>>((const float4*)z, ssum, (float4*)out, ND4);
}